// GateRow_601295422061
// MI455X (gfx1250) — compile-verified
//
#include <hip/hip_runtime.h>
#include <hip/hip_bf16.h>
#include <stdint.h>

// Problem constants (match reference).
constexpr int BATCH   = 2048;
constexpr int N_GATES = 65536;
constexpr int N_IN    = 65536;
constexpr int WB      = BATCH / 32;  // 64 packed batch-words per input column
constexpr int GPT     = 4;           // gates per thread (float4 output stores)
constexpr int GTILE   = 256 * GPT;   // 1024 gates per block
constexpr int WSPLIT  = 4;           // w-dimension split across blockIdx.y
constexpr int WPB     = WB / WSPLIT; // 16 batch-words per gate block
constexpr int WQ      = WPB / 4;     // 4 b128 gather steps (4 words each)

typedef int      v4i __attribute__((ext_vector_type(4)));
typedef unsigned v4u __attribute__((ext_vector_type(4)));
typedef float    v4f __attribute__((ext_vector_type(4)));

// Generic pointer to a __shared__ object -> 32-bit LDS byte offset
// (addrspace(3) ptrtoint), used as the VDST operand of the CDNA5
// GLOBAL_LOAD_ASYNC_TO_LDS instructions.
__device__ __forceinline__ unsigned lds_off(const void* p) {
  return (unsigned)(unsigned long long)
      (__attribute__((address_space(3))) const void*)(p);
}

// ---------------------------------------------------------------------------
// Kernel 1: bit-pack x across the batch dimension, TRANSPOSED layout:
//   yT[i * WB + w] bit k == (x[(w*32+k) * N_IN + i] & 1)
// so each input column's 64 batch-words are one contiguous, 256B-aligned
// block (2 cache lines) -> the gate kernel's gathers get full line use.
// One thread per input column i; every load is 128B coalesced across the
// wave (lanes = consecutive i). x is streamed once -> non-temporal loads.
// ---------------------------------------------------------------------------
__global__ __launch_bounds__(256) void pack_kernel(const int* __restrict__ x,
                                                   unsigned* __restrict__ yT) {
  const int i = blockIdx.x * 256 + threadIdx.x;   // input column
  for (int c = 0; c < WB / 4; ++c) {              // 16 chunks of 128 rows
    unsigned wbuf[4];
#pragma unroll
    for (int q = 0; q < 4; ++q) {
      unsigned bits = 0;
#pragma unroll
      for (int k = 0; k < 32; ++k) {
        const int b = c * 128 + q * 32 + k;       // batch row
        const int v = __builtin_nontemporal_load(x + (size_t)b * N_IN + i);
        bits |= (unsigned)(v & 1) << k;
      }
      wbuf[q] = bits;
    }
    // Reused by the gate kernel -> regular-temporal (L2 resident), aligned.
    *(uint4*)(yT + (size_t)i * WB + c * 4) =
        make_uint4(wbuf[0], wbuf[1], wbuf[2], wbuf[3]);
  }
}

// ---------------------------------------------------------------------------
// Kernel 2: evaluate gates 128 batch rows at a time per gather step.
// Each thread owns 4 consecutive gates. Metadata staged via the CDNA5
// async-to-LDS path (GLOBAL_LOAD_ASYNC_TO_LDS_B128 + s_wait_asynccnt).
// Gathers: one b128 per column per step reads 4 contiguous batch-words of
// yT's transposed layout -> 16B fully used, lines walked sequentially across
// steps; total gather traffic ~32MB, all L2-resident (table is 16MB).
// Output: NT float4 stores, 512B contiguous per wave per row.
// ---------------------------------------------------------------------------
__global__ __launch_bounds__(256) void gate_kernel(
    const unsigned* __restrict__ yT,
    const unsigned char* __restrict__ gates,   // numpy bool: 1B/entry, 4/gate
    const int* __restrict__ choices,           // 2 ints/gate
    float* __restrict__ out) {                 // [BATCH, N_GATES]
  __shared__ int      s_ch[GTILE * 2];  // 8 KB
  __shared__ unsigned s_tt[GTILE];      // 4 KB (one u32 = 4 tt bytes per gate)

  const int t  = threadIdx.x;
  const int g0 = blockIdx.x * GTILE + t * GPT;   // first of this thread's gates

  {
    // 32B of choices for 4 gates -> two async b128 DMAs into private slots.
    const unsigned ldsc = lds_off(&s_ch[t * 2 * GPT]);
    const unsigned vofc = (unsigned)g0 * 8u;
    asm volatile("global_load_async_to_lds_b128 %0, %1, %2"
                 :: "v"(ldsc), "v"(vofc), "s"(choices) : "memory");
    asm volatile("global_load_async_to_lds_b128 %0, %1, %2"
                 :: "v"(ldsc + 16u), "v"(vofc + 16u), "s"(choices) : "memory");
    // 16B of truth tables for 4 gates -> one async b128 DMA.
    const unsigned ldst = lds_off(&s_tt[t * GPT]);
    const unsigned voft = (unsigned)g0 * 4u;
    asm volatile("global_load_async_to_lds_b128 %0, %1, %2"
                 :: "v"(ldst), "v"(voft), "s"(gates) : "memory");
    asm volatile("s_wait_asynccnt 0" ::: "memory");
  }

  int      c0[GPT], c1[GPT];
  unsigned m0[GPT], m1[GPT], m2[GPT], m3[GPT];
#pragma unroll
  for (int j = 0; j < GPT; ++j) {
    c0[j] = s_ch[t * 2 * GPT + 2 * j];
    c1[j] = s_ch[t * 2 * GPT + 2 * j + 1];
    const unsigned ttw = s_tt[t * GPT + j];
    // idx = 2*bit(c0) + bit(c1); tt byte idx -> all-ones/zero word mask.
    m0[j] = (ttw & 0x000000FFu) ? 0xFFFFFFFFu : 0u;  // a=0,c=0
    m1[j] = (ttw & 0x0000FF00u) ? 0xFFFFFFFFu : 0u;  // a=0,c=1
    m2[j] = (ttw & 0x00FF0000u) ? 0xFFFFFFFFu : 0u;  // a=1,c=0
    m3[j] = (ttw & 0xFF000000u) ? 0xFFFFFFFFu : 0u;  // a=1,c=1
  }

  const int wbase = blockIdx.y * WPB;
  for (int wq = 0; wq < WQ; ++wq) {
    const int w0 = wbase + wq * 4;                 // 4 batch-words this step

    v4u r4[GPT];
#pragma unroll
    for (int j = 0; j < GPT; ++j) {
      const v4u a = *(const v4u*)(yT + (size_t)c0[j] * WB + w0);
      const v4u c = *(const v4u*)(yT + (size_t)c1[j] * WB + w0);
      r4[j] = ((~a & ~c) & m0[j]) | ((~a & c) & m1[j]) |
              (( a & ~c) & m2[j]) | (( a &  c) & m3[j]);
    }

#pragma unroll
    for (int d = 0; d < 4; ++d) {                  // word within step
      float* obase = out + (size_t)((w0 + d) * 32) * N_GATES + g0;
#pragma unroll
      for (int k = 0; k < 32; ++k) {               // batch row within word
        v4f f;
        f.x = ((r4[0][d] >> k) & 1u) ? 1.0f : 0.0f;
        f.y = ((r4[1][d] >> k) & 1u) ? 1.0f : 0.0f;
        f.z = ((r4[2][d] >> k) & 1u) ? 1.0f : 0.0f;
        f.w = ((r4[3][d] >> k) & 1u) ? 1.0f : 0.0f;
        __builtin_nontemporal_store(f, (v4f*)(obase + (size_t)k * N_GATES));
      }
    }
  }
}

// ---------------------------------------------------------------------------
// d_in[0] = x       int32  [BATCH * N_IN]
// d_in[1] = gates   bool   [N_GATES * 4]   (1 byte per element)
// d_in[2] = choices int32  [N_GATES * 2]
// d_out   = float   [BATCH * N_GATES]
// d_ws    : needs 16 MB for the packed table yT.
// ---------------------------------------------------------------------------
extern "C" void kernel_launch(void* const* d_in, const int* in_sizes, int n_in,
                              void* d_out, int out_size, void* d_ws, size_t ws_size,
                              hipStream_t stream) {
  const int*           x       = (const int*)d_in[0];
  const unsigned char* gates   = (const unsigned char*)d_in[1];
  const int*           choices = (const int*)d_in[2];
  float*               out     = (float*)d_out;
  unsigned*            yT      = (unsigned*)d_ws;   // N_IN * WB words = 16 MB

  dim3 pblk(256), pgrid(N_IN / 256);                     // 256 blocks
  pack_kernel<<<pgrid, pblk, 0, stream>>>(x, yT);

  dim3 gblk(256), ggrid(N_GATES / GTILE, WSPLIT);        // (64, 4)
  gate_kernel<<<ggrid, gblk, 0, stream>>>(yT, gates, choices, out);
}